// GCN_53893249630532
// MI455X (gfx1250) — compile-verified
//
#include <hip/hip_runtime.h>

#define NNODES 20000
#define NEDGES 320000
#define NGRAPH 64
#define HD 256

typedef __attribute__((ext_vector_type(2))) float v2f;
typedef __attribute__((ext_vector_type(8))) float v8f;

// ---------------- elementwise utilities ----------------

__global__ void k_zero(float* __restrict__ p, int n) {
  int i = blockIdx.x * blockDim.x + threadIdx.x;
  if (i < n) p[i] = 0.f;
}

__global__ void k_copy_neg(float* __restrict__ dst, const float* __restrict__ src, int n) {
  int i = blockIdx.x * blockDim.x + threadIdx.x;
  if (i < n) dst[i] = -src[i];
}

__global__ void k_neg_inplace(float* __restrict__ p, int n) {
  int i = blockIdx.x * blockDim.x + threadIdx.x;
  if (i < n) p[i] = -p[i];
}

__global__ void k_relu_copy(float* __restrict__ dst, const float* __restrict__ src, int n) {
  int i = blockIdx.x * blockDim.x + threadIdx.x;
  if (i < n) dst[i] = fmaxf(src[i], 0.f);
}

// acc[n][c] = b[c]  (pre-seed GEMM accumulator with bias)
__global__ void k_init_bias(float* __restrict__ acc, const float* __restrict__ b, int n) {
  int i = blockIdx.x * blockDim.x + threadIdx.x;
  if (i < n) acc[i] = b[i & (HD - 1)];
}

// ---------------- graph normalization ----------------

__global__ void k_degree(const int* __restrict__ row, float* __restrict__ deg) {
  int e = blockIdx.x * blockDim.x + threadIdx.x;
  if (e < NEDGES) unsafeAtomicAdd(&deg[row[e]], 1.0f);
}

__global__ void k_dinv(float* __restrict__ d) {
  int i = blockIdx.x * blockDim.x + threadIdx.x;
  if (i < NNODES) {
    float v = d[i];
    d[i] = (v > 0.f) ? rsqrtf(fmaxf(v, 1.f)) : 0.f;
  }
}

__global__ void k_norm(const int* __restrict__ row, const int* __restrict__ col,
                       const float* __restrict__ dinv, float* __restrict__ norm) {
  int e = blockIdx.x * blockDim.x + threadIdx.x;
  if (e < NEDGES) norm[e] = dinv[row[e]] * dinv[col[e]];
}

// ---------------- Lhat scatter:  y[row[e]] += coef * norm[e] * v[col[e]] ----------------

__global__ void k_scatter6(float* __restrict__ y, const float* __restrict__ v,
                           const int* __restrict__ row, const int* __restrict__ col,
                           const float* __restrict__ norm, float coef) {
  int idx = blockIdx.x * blockDim.x + threadIdx.x;
  if (idx >= NEDGES * 6) return;
  int e = idx / 6, f = idx - e * 6;
  float w = coef * norm[e];
  unsafeAtomicAdd(&y[row[e] * 6 + f], w * v[col[e] * 6 + f]);
}

__global__ void k_scatter256(float* __restrict__ y, const float* __restrict__ v,
                             const int* __restrict__ row, const int* __restrict__ col,
                             const float* __restrict__ norm, float coef) {
  long long idx = (long long)blockIdx.x * blockDim.x + threadIdx.x;
  if (idx >= (long long)NEDGES * 64) return;
  int e = (int)(idx >> 6);
  int c4 = ((int)idx & 63) << 2;
  float w = coef * norm[e];
  const float4 src = *reinterpret_cast<const float4*>(v + (long long)col[e] * HD + c4);
  float* dst = y + (long long)row[e] * HD + c4;
  unsafeAtomicAdd(dst + 0, w * src.x);
  unsafeAtomicAdd(dst + 1, w * src.y);
  unsafeAtomicAdd(dst + 2, w * src.z);
  unsafeAtomicAdd(dst + 3, w * src.w);
}

// ---------------- layer-1 small GEMM (K=6): C[20000x256] += A[20000x6] @ W[6x256] ----------------

__global__ void k_gemm6(const float* __restrict__ A, const float* __restrict__ W,
                        float* __restrict__ C) {
  int idx = blockIdx.x * blockDim.x + threadIdx.x;
  if (idx >= NNODES * HD) return;
  int n = idx >> 8, c = idx & (HD - 1);
  float s = 0.f;
#pragma unroll
  for (int j = 0; j < 6; ++j) s += A[n * 6 + j] * W[j * HD + c];
  C[idx] += s;
}

// ---------------- main f32 WMMA GEMM: C[Mx256] += A[Mx256] @ B[256x256] ----------------
// block = 4 waves (128 threads); block owns a 16-row stripe; wave w owns cols [64w, 64w+64)
// as four 16x16 f32 accumulators. Uses V_WMMA_F32_16X16X4_F32.
// A frag (16x4 f32): lane m in half 0 holds K=kb,kb+1; half 1 holds K=kb+2,kb+3 (ISA 7.12.2).
// B frag (4x16 f32): lanes hold N; same half-split over K as A (symmetric layout).
// C/D (16x16 f32 in 8 VGPRs... here 8-elt vector): c[v] = C[row0 + v + 8*half][col + (lane&15)].

__global__ __launch_bounds__(128) void k_gemm_wmma(const float* __restrict__ A,
                                                   const float* __restrict__ B,
                                                   float* __restrict__ C) {
  const int lane = threadIdx.x & 31;
  const int wave = threadIdx.x >> 5;
  const int half = lane >> 4;
  const int m    = lane & 15;
  const int row0 = blockIdx.x * 16;
  const int col0 = wave * 64;

  // accumulate into existing C
  float* cbase = C + (long long)(row0 + 8 * half) * HD + col0 + m;
  v8f c0, c1, c2, c3;
#pragma unroll
  for (int v = 0; v < 8; ++v) {
    c0[v] = cbase[v * HD + 0];
    c1[v] = cbase[v * HD + 16];
    c2[v] = cbase[v * HD + 32];
    c3[v] = cbase[v * HD + 48];
  }

  const float* arow = A + (long long)(row0 + m) * HD;
  for (int kb = 0; kb < HD; kb += 4) {
    const int ka = kb + 2 * half;
    v2f a;
    a.x = arow[ka];
    a.y = arow[ka + 1];
    const float* bp = B + (long long)ka * HD + col0 + m;
    v2f b0, b1, b2, b3;
    b0.x = bp[0];  b0.y = bp[HD + 0];
    b1.x = bp[16]; b1.y = bp[HD + 16];
    b2.x = bp[32]; b2.y = bp[HD + 32];
    b3.x = bp[48]; b3.y = bp[HD + 48];
    c0 = __builtin_amdgcn_wmma_f32_16x16x4_f32(false, a, false, b0, (short)0, c0, false, false);
    c1 = __builtin_amdgcn_wmma_f32_16x16x4_f32(false, a, false, b1, (short)0, c1, false, false);
    c2 = __builtin_amdgcn_wmma_f32_16x16x4_f32(false, a, false, b2, (short)0, c2, false, false);
    c3 = __builtin_amdgcn_wmma_f32_16x16x4_f32(false, a, false, b3, (short)0, c3, false, false);
  }

#pragma unroll
  for (int v = 0; v < 8; ++v) {
    cbase[v * HD + 0]  = c0[v];
    cbase[v * HD + 16] = c1[v];
    cbase[v * HD + 32] = c2[v];
    cbase[v * HD + 48] = c3[v];
  }
}

// ---------------- pooling: g[batch[n]][c] = max over nodes of relu(h[n][c]) ----------------
// relu output >= 0, pool buffer zero-initialized -> int-bit atomicMax is order-preserving.

__global__ void k_pool(const float* __restrict__ h, const int* __restrict__ batch,
                       float* __restrict__ g) {
  int idx = blockIdx.x * blockDim.x + threadIdx.x;
  if (idx >= NNODES * HD) return;
  int n = idx >> 8, c = idx & (HD - 1);
  float val = fmaxf(h[idx], 0.f);
  atomicMax(reinterpret_cast<int*>(g + batch[n] * HD + c), __float_as_int(val));
}

// ---------------- small dense head: Y[MxN] = act(A[MxK] @ W[KxN] + b) ----------------

__global__ void k_fc(const float* __restrict__ A, const float* __restrict__ W,
                     const float* __restrict__ b, float* __restrict__ Y,
                     int M, int K, int N, int relu) {
  int idx = blockIdx.x * blockDim.x + threadIdx.x;
  if (idx >= M * N) return;
  int mm = idx / N, nn = idx - mm * N;
  float s = b[nn];
  for (int k = 0; k < K; ++k) s += A[mm * K + k] * W[k * N + nn];
  Y[idx] = relu ? fmaxf(s, 0.f) : s;
}

// ---------------- host orchestration ----------------

static inline dim3 gsz(long long n, int bs) { return dim3((unsigned)((n + bs - 1) / bs)); }

extern "C" void kernel_launch(void* const* d_in, const int* in_sizes, int n_in,
                              void* d_out, int out_size, void* d_ws, size_t ws_size,
                              hipStream_t stream) {
  const float* x    = (const float*)d_in[0];
  const int*   ei   = (const int*)d_in[1];
  const int*   row  = ei;
  const int*   col  = ei + NEDGES;
  const int*   batch= (const int*)d_in[2];
  const float* W1   = (const float*)d_in[3];
  const float* b1   = (const float*)d_in[4];
  const float* W2   = (const float*)d_in[5];
  const float* b2   = (const float*)d_in[6];
  const float* W3   = (const float*)d_in[7];
  const float* b3   = (const float*)d_in[8];
  const float* f1w  = (const float*)d_in[9];
  const float* f1b  = (const float*)d_in[10];
  const float* f2w  = (const float*)d_in[11];
  const float* f2b  = (const float*)d_in[12];
  const float* lw   = (const float*)d_in[13];
  const float* lb   = (const float*)d_in[14];
  float* out = (float*)d_out;

  float* w = (float*)d_ws;
  float* dinv = w;  w += NNODES;
  float* norm = w;  w += NEDGES;
  float* T1a  = w;  w += NNODES * 6;
  float* T2a  = w;  w += NNODES * 6;
  float* H    = w;  w += (long long)NNODES * HD;
  float* T1   = w;  w += (long long)NNODES * HD;
  float* ACC  = w;  w += (long long)NNODES * HD;
  float* G    = w;  w += NGRAPH * HD;
  float* FC1o = w;  w += NGRAPH * 512;
  float* FC2o = w;  w += NGRAPH * HD;

  const int BS = 256;
  const long long NH = (long long)NNODES * HD;     // 5,120,000
  const long long E4 = (long long)NEDGES * 64;     // 20,480,000

  // --- symmetric normalization ---
  k_zero<<<gsz(NNODES, BS), BS, 0, stream>>>(dinv, NNODES);
  k_degree<<<gsz(NEDGES, BS), BS, 0, stream>>>(row, dinv);
  k_dinv<<<gsz(NNODES, BS), BS, 0, stream>>>(dinv);
  k_norm<<<gsz(NEDGES, BS), BS, 0, stream>>>(row, col, dinv, norm);

  // --- layer 1 (K=3, in=6) ---
  k_init_bias<<<gsz(NH, BS), BS, 0, stream>>>(ACC, b1, (int)NH);
  k_gemm6<<<gsz(NH, BS), BS, 0, stream>>>(x, W1 + 0 * 6 * HD, ACC);
  k_zero<<<gsz(NNODES * 6, BS), BS, 0, stream>>>(T1a, NNODES * 6);
  k_scatter6<<<gsz((long long)NEDGES * 6, BS), BS, 0, stream>>>(T1a, x, row, col, norm, -1.f);
  k_gemm6<<<gsz(NH, BS), BS, 0, stream>>>(T1a, W1 + 1 * 6 * HD, ACC);
  k_copy_neg<<<gsz(NNODES * 6, BS), BS, 0, stream>>>(T2a, x, NNODES * 6);
  k_scatter6<<<gsz((long long)NEDGES * 6, BS), BS, 0, stream>>>(T2a, T1a, row, col, norm, -2.f);
  k_gemm6<<<gsz(NH, BS), BS, 0, stream>>>(T2a, W1 + 2 * 6 * HD, ACC);
  k_relu_copy<<<gsz(NH, BS), BS, 0, stream>>>(H, ACC, (int)NH);

  const int GB = NNODES / 16;  // 1250 row-stripe blocks for WMMA GEMM
  const long long KW = (long long)HD * HD;

  // --- layer 2 (K=4, 256->256) ---
  k_init_bias<<<gsz(NH, BS), BS, 0, stream>>>(ACC, b2, (int)NH);
  k_gemm_wmma<<<GB, 128, 0, stream>>>(H, W2 + 0 * KW, ACC);
  k_zero<<<gsz(NH, BS), BS, 0, stream>>>(T1, (int)NH);
  k_scatter256<<<gsz(E4, BS), BS, 0, stream>>>(T1, H, row, col, norm, -1.f);
  k_gemm_wmma<<<GB, 128, 0, stream>>>(T1, W2 + 1 * KW, ACC);
  k_neg_inplace<<<gsz(NH, BS), BS, 0, stream>>>(H, (int)NH);
  k_scatter256<<<gsz(E4, BS), BS, 0, stream>>>(H, T1, row, col, norm, -2.f);
  k_gemm_wmma<<<GB, 128, 0, stream>>>(H, W2 + 2 * KW, ACC);
  k_neg_inplace<<<gsz(NH, BS), BS, 0, stream>>>(T1, (int)NH);
  k_scatter256<<<gsz(E4, BS), BS, 0, stream>>>(T1, H, row, col, norm, -2.f);
  k_gemm_wmma<<<GB, 128, 0, stream>>>(T1, W2 + 3 * KW, ACC);
  k_relu_copy<<<gsz(NH, BS), BS, 0, stream>>>(H, ACC, (int)NH);

  // --- layer 3 (K=5, 256->256) ---
  k_init_bias<<<gsz(NH, BS), BS, 0, stream>>>(ACC, b3, (int)NH);
  k_gemm_wmma<<<GB, 128, 0, stream>>>(H, W3 + 0 * KW, ACC);
  k_zero<<<gsz(NH, BS), BS, 0, stream>>>(T1, (int)NH);
  k_scatter256<<<gsz(E4, BS), BS, 0, stream>>>(T1, H, row, col, norm, -1.f);
  k_gemm_wmma<<<GB, 128, 0, stream>>>(T1, W3 + 1 * KW, ACC);
  k_neg_inplace<<<gsz(NH, BS), BS, 0, stream>>>(H, (int)NH);
  k_scatter256<<<gsz(E4, BS), BS, 0, stream>>>(H, T1, row, col, norm, -2.f);
  k_gemm_wmma<<<GB, 128, 0, stream>>>(H, W3 + 2 * KW, ACC);
  k_neg_inplace<<<gsz(NH, BS), BS, 0, stream>>>(T1, (int)NH);
  k_scatter256<<<gsz(E4, BS), BS, 0, stream>>>(T1, H, row, col, norm, -2.f);
  k_gemm_wmma<<<GB, 128, 0, stream>>>(T1, W3 + 3 * KW, ACC);
  k_neg_inplace<<<gsz(NH, BS), BS, 0, stream>>>(H, (int)NH);
  k_scatter256<<<gsz(E4, BS), BS, 0, stream>>>(H, T1, row, col, norm, -2.f);
  k_gemm_wmma<<<GB, 128, 0, stream>>>(H, W3 + 4 * KW, ACC);

  // --- global max pool (relu fused) ---
  k_zero<<<gsz(NGRAPH * HD, BS), BS, 0, stream>>>(G, NGRAPH * HD);
  k_pool<<<gsz(NH, BS), BS, 0, stream>>>(ACC, batch, G);

  // --- MLP head ---
  k_fc<<<gsz(NGRAPH * 512, BS), BS, 0, stream>>>(G, f1w, f1b, FC1o, NGRAPH, 256, 512, 1);
  k_fc<<<gsz(NGRAPH * 256, BS), BS, 0, stream>>>(FC1o, f2w, f2b, FC2o, NGRAPH, 512, 256, 1);
  k_fc<<<gsz(NGRAPH * 24, BS), BS, 0, stream>>>(FC2o, lw, lb, out, NGRAPH, 256, 24, 0);
}